// SLSTM_44933947851219
// MI455X (gfx1250) — compile-verified
//
#include <hip/hip_runtime.h>
#include <math.h>

typedef float v2f __attribute__((ext_vector_type(2)));
typedef float v8f __attribute__((ext_vector_type(8)));

#define Bn   16
#define Tn   2048
#define In   512
#define Hn   512
#define G4H  2048     // 4*H
#define NWG  32       // phase-2 workgroups (each owns 16 h-columns)
#define LPAD 516      // padded LDS row stride (floats): bank stride 4 -> conflict-free

__device__ __forceinline__ float sigmoid_f(float v) {
    return 1.0f / (1.0f + __expf(-v));
}
__device__ __forceinline__ float tanh_f(float v) {
    float a = fabsf(v);
    float e = __expf(-2.0f * a);
    float t = (1.0f - e) / (1.0f + e);
    return copysignf(t, v);
}

// ---------------------------------------------------------------------------
// Phase 0: reset grid-sync counter (workspace is not re-poisoned by harness)
// ---------------------------------------------------------------------------
__global__ void SLSTM_zero_sync(int* p) {
    if (threadIdx.x == 0)
        __hip_atomic_store(p, 0, __ATOMIC_RELEASE, __HIP_MEMORY_SCOPE_AGENT);
}

// ---------------------------------------------------------------------------
// Phase 1: xg[t][b][g] = sum_k x[b][t][k]*Wi[g][k] + bi[g] + bh[g]
// Register-blocked: each wave computes 4 t-tiles x 2 n-tiles (8 accumulators).
// A (16x4 f32): lanes 0-15 = M(batch), v0/v1 = K (lanes 16-31 hold K+2/K+3)
// B (4x16 f32): lanes 0-15 = N,        v0/v1 = K (lanes 16-31 hold K+2/K+3)
// ---------------------------------------------------------------------------
__global__ __launch_bounds__(256) void SLSTM_xgemm(
    const float* __restrict__ x, const float* __restrict__ Wi,
    const float* __restrict__ bi, const float* __restrict__ bh,
    float* __restrict__ xg)
{
    const int lane = threadIdx.x & 31;
    const int wave = threadIdx.x >> 5;                    // 0..7
    const int t0   = (blockIdx.x >> 3) * 4;               // 512 t-groups of 4
    const int nt0  = ((blockIdx.x & 7) * 8 + wave) * 2;   // even tile id, 0..126
    const int n0   = nt0 * 16;
    const int bL   = lane & 15;
    const int kh   = lane >> 4;                           // 0/1 (K half)

    const float* a0 = x  + ((size_t)bL * Tn + t0) * In + 2 * kh;  // + i*In per t
    const float* b0 = Wi + (size_t)(n0 + bL) * In + 2 * kh;       // + j*16*In per ntile

    v8f c[4][2] = {};
    for (int k = 0; k < In; k += 4) {
        v2f a[4], b[2];
#pragma unroll
        for (int i = 0; i < 4; ++i)
            a[i] = *(const v2f*)(a0 + (size_t)i * In + k);
#pragma unroll
        for (int j = 0; j < 2; ++j)
            b[j] = *(const v2f*)(b0 + (size_t)j * 16 * In + k);
#pragma unroll
        for (int i = 0; i < 4; ++i)
#pragma unroll
            for (int j = 0; j < 2; ++j)
                c[i][j] = __builtin_amdgcn_wmma_f32_16x16x4_f32(
                    false, a[i], false, b[j], (short)0, c[i][j], false, false);
    }
#pragma unroll
    for (int j = 0; j < 2; ++j) {
        const float bias = bi[n0 + 16 * j + bL] + bh[n0 + 16 * j + bL];
#pragma unroll
        for (int i = 0; i < 4; ++i) {
            // C/D layout: VGPR v, lanes 0-15 (M=v,N=lane); lanes 16-31 (M=v+8,N=lane-16)
            float* orow = xg + ((size_t)(t0 + i) * 16 + 8 * kh) * G4H + n0 + 16 * j + bL;
#pragma unroll
            for (int v = 0; v < 8; ++v)
                orow[(size_t)v * G4H] = c[i][j][v] + bias;
        }
    }
}

// ---------------------------------------------------------------------------
// Phase 2: persistent scan. 32 WGs x 256 threads (8 waves).
// Wave wv: gate g = wv&3, K-half ks = wv>>2 (split-K halves the WMMA chain).
// h staged into LDS via CDNA5 async global->LDS loads (ASYNCcnt).
// xg tile for t+1 prefetched into registers while the grid sync settles.
// ---------------------------------------------------------------------------
__global__ __launch_bounds__(256) void SLSTM_scan(
    const float* __restrict__ xg, const float* __restrict__ Wh,
    const float* __restrict__ h0, const float* __restrict__ c0,
    float* __restrict__ hbuf, int* __restrict__ syncp,
    float* __restrict__ out, float* __restrict__ hT, float* __restrict__ cT)
{
    __shared__ float Whs[64 * LPAD];     // owned Wh rows, resident for all steps
    __shared__ float hls[16 * LPAD];     // staged h [b][k]
    __shared__ float gl[8 * 272];        // per-wave partial gate tiles (16x17 each)
    __shared__ float cls[256];           // cell state for owned columns

    const int tid  = threadIdx.x;
    const int lane = tid & 31;
    const int wv   = tid >> 5;           // 0..7
    const int g    = wv & 3;             // gate (i,f,g,o)
    const int ks   = wv >> 2;            // K half
    const int kb   = ks * 256;           // K base
    const int w    = blockIdx.x;         // 0..31
    const int j0   = w * 16;             // owned h-column base
    const int bL   = lane & 15;
    const int kh   = lane >> 4;

    // --- stage owned Wh slice into LDS (once): 64 rows x 512, float4 loads ---
    for (int i = tid; i < 64 * 128; i += 256) {
        int lr = i >> 7;                 // local row 0..63
        int kc = (i & 127) << 2;
        int gg = lr >> 4;
        int rr = lr & 15;
        float4 v = *(const float4*)(Wh + (size_t)(gg * Hn + j0 + rr) * Hn + kc);
        *(float4*)(&Whs[lr * LPAD + kc]) = v;
    }
    // --- init c (LDS) and h buffer 0 (global) for owned columns ---
    {
        int b = tid >> 4, j = tid & 15;
        cls[tid] = c0[b * Hn + j0 + j];
        hbuf[b * Hn + j0 + j] = h0[b * Hn + j0 + j];
    }
    __syncthreads();

    int epoch = 1;
    if (tid == 0) {
        __hip_atomic_fetch_add(syncp, 1, __ATOMIC_RELEASE, __HIP_MEMORY_SCOPE_AGENT);
        while (__hip_atomic_load(syncp, __ATOMIC_ACQUIRE, __HIP_MEMORY_SCOPE_AGENT)
               < epoch * NWG)
            __builtin_amdgcn_s_sleep(1);
    }
    __syncthreads();

    // xg prefetch registers (only ks==0 waves accumulate the xg/bias term)
    float xv[8];
    if (ks == 0) {
        const float* xrow = xg + (size_t)(8 * kh) * G4H + g * Hn + j0 + bL;
#pragma unroll
        for (int v = 0; v < 8; ++v) xv[v] = xrow[(size_t)v * G4H];
    }

    const unsigned lbase = (unsigned)(uintptr_t)(&hls[0]);

    for (int t = 0; t < Tn; ++t) {
        const float* hsrc = hbuf + (size_t)(t & 1) * (Bn * Hn);
        float*       hdst = hbuf + (size_t)((t + 1) & 1) * (Bn * Hn);

        // stage full h [16][512] -> LDS via async global->LDS (ASYNCcnt path)
#pragma unroll
        for (int r = 0; r < 8; ++r) {
            int i  = tid + r * 256;          // float4 index, 2048 total
            int b  = i >> 7;
            int kc = (i & 127) << 2;
            unsigned loff = lbase + (unsigned)((b * LPAD + kc) * 4);
            unsigned goff = (unsigned)((b * Hn + kc) * 4);
            asm volatile("global_load_async_to_lds_b128 %0, %1, %2"
                         :: "v"(loff), "v"(goff), "s"(hsrc) : "memory");
        }
        asm volatile("s_wait_asynccnt 0x0" ::: "memory");
        __syncthreads();

        // K-half GEMM: 64 WMMAs, A from hls, B from Whs (bank-conflict-free)
        v8f c = {};
        if (ks == 0) {
#pragma unroll
            for (int v = 0; v < 8; ++v) c[v] = xv[v];
        }
        const float* ap = &hls[bL * LPAD + kb + 2 * kh];
        const float* bp = &Whs[(g * 16 + bL) * LPAD + kb + 2 * kh];
        for (int k = 0; k < 256; k += 4) {
            v2f a = *(const v2f*)(ap + k);
            v2f b = *(const v2f*)(bp + k);
            c = __builtin_amdgcn_wmma_f32_16x16x4_f32(false, a, false, b,
                                                      (short)0, c, false, false);
        }
#pragma unroll
        for (int v = 0; v < 8; ++v)
            gl[wv * 272 + (v + 8 * kh) * 17 + bL] = c[v];
        __syncthreads();

        // elementwise LSTM update: one (b,j) per thread; sum split-K partials
        {
            int b = tid >> 4, j = tid & 15;
            float pi = gl[0 * 272 + b * 17 + j] + gl[4 * 272 + b * 17 + j];
            float pf = gl[1 * 272 + b * 17 + j] + gl[5 * 272 + b * 17 + j];
            float pg = gl[2 * 272 + b * 17 + j] + gl[6 * 272 + b * 17 + j];
            float po = gl[3 * 272 + b * 17 + j] + gl[7 * 272 + b * 17 + j];
            float ig = sigmoid_f(pi);
            float fg = sigmoid_f(pf);
            float gv = tanh_f(pg);
            float og = sigmoid_f(po);
            float cn = fg * cls[tid] + ig * gv;
            float hn = og * tanh_f(cn);
            cls[tid] = cn;
            hdst[b * Hn + j0 + j] = hn;
            out[((size_t)b * Tn + t) * Hn + j0 + j] = hn;
            if (t == Tn - 1) {
                hT[b * Hn + j0 + j] = hn;
                cT[b * Hn + j0 + j] = cn;
            }
        }
        __syncthreads();   // protect gl/hls before next step overwrites them

        // grid-wide sync; overlap xg(t+1) prefetch with the sync round-trip
        epoch++;
        if (tid == 0)
            __hip_atomic_fetch_add(syncp, 1, __ATOMIC_RELEASE,
                                   __HIP_MEMORY_SCOPE_AGENT);
        if (ks == 0 && t + 1 < Tn) {
            const float* xrow = xg + ((size_t)(t + 1) * 16 + 8 * kh) * G4H
                                   + g * Hn + j0 + bL;
#pragma unroll
            for (int v = 0; v < 8; ++v) xv[v] = xrow[(size_t)v * G4H];
        }
        if (tid == 0) {
            while (__hip_atomic_load(syncp, __ATOMIC_ACQUIRE,
                                     __HIP_MEMORY_SCOPE_AGENT) < epoch * NWG)
                __builtin_amdgcn_s_sleep(1);
        }
        __syncthreads();
    }
}

// ---------------------------------------------------------------------------
extern "C" void kernel_launch(void* const* d_in, const int* in_sizes, int n_in,
                              void* d_out, int out_size, void* d_ws, size_t ws_size,
                              hipStream_t stream) {
    const float* x  = (const float*)d_in[0];
    const float* h0 = (const float*)d_in[1];
    const float* c0 = (const float*)d_in[2];
    const float* Wi = (const float*)d_in[3];
    const float* bi = (const float*)d_in[4];
    const float* Wh = (const float*)d_in[5];
    const float* bh = (const float*)d_in[6];

    float* out = (float*)d_out;                      // [B,T,H]
    float* hT  = out + (size_t)Bn * Tn * Hn;         // [B,H]
    float* cT  = hT + (size_t)Bn * Hn;               // [B,H]

    char* ws    = (char*)d_ws;
    int*  syncp = (int*)ws;                          // [0,256): sync counter
    float* hbuf = (float*)(ws + 256);                // 2*16*512 floats (64 KB)
    float* xg   = (float*)(ws + 131072);             // [T,16,4H] fp32 (256 MB)

    SLSTM_zero_sync<<<1, 64, 0, stream>>>(syncp);
    SLSTM_xgemm<<<512 * 8, 256, 0, stream>>>(x, Wi, bi, bh, xg);
    SLSTM_scan<<<NWG, 256, 0, stream>>>(xg, Wh, h0, c0, hbuf, syncp, out, hT, cT);
}